// SimplifiedSSM_37134287241953
// MI455X (gfx1250) — compile-verified
//
#include <hip/hip_runtime.h>
#include <hip/hip_bf16.h>
#include <math.h>

// ---------------------------------------------------------------------------
// SimplifiedSSM on MI455X (gfx1250, wave32, WMMA)
//   Phase 1: bx = x @ B_w^T            (f32 WMMA GEMM, async-LDS double buffer)
//   Phase 2: h_t = tanh(h_{t-1}A^T+bx) (A-slice resident in 261KB LDS/WGP,
//                                       cross-WGP step sync via L2 counters,
//                                       s_cluster_barrier fast path)
//   Phase 3: y  = H @ C_w^T + D*x      (f32 WMMA GEMM with fused epilogue)
// ---------------------------------------------------------------------------

#define DIM   1024
#define BSZ   4
#define LSEQ  4096
#define NROWS (BSZ * LSEQ)      // 16384
#define LDT   36                // padded LDS tile row stride (floats): 16B-aligned
                                // rows AND conflict-free (36*m mod 64, m=0..15)

typedef __attribute__((ext_vector_type(2))) float v2f;
typedef __attribute__((ext_vector_type(8))) float v8f;
typedef __attribute__((__vector_size__(4 * sizeof(int)))) int v4i_t;

// --------------------------- async global->LDS -----------------------------
#if __has_builtin(__builtin_amdgcn_global_load_async_to_lds_b128)
#define HAVE_ASYNC_LDS 1
__device__ __forceinline__ void async_b128(const float* g, float* lds) {
    __builtin_amdgcn_global_load_async_to_lds_b128(
        (__attribute__((address_space(1))) v4i_t*)(g),
        (__attribute__((address_space(3))) v4i_t*)(lds), 0, 0);
}
__device__ __forceinline__ void wait_async0() {
#if __has_builtin(__builtin_amdgcn_s_wait_asynccnt)
    __builtin_amdgcn_s_wait_asynccnt(0);
#else
    asm volatile("s_wait_asynccnt 0x0" ::: "memory");
#endif
}
#else
#define HAVE_ASYNC_LDS 0
__device__ __forceinline__ void async_b128(const float* g, float* lds) {
    float4 v = *(const float4*)g;       // synchronous fallback
    lds[0] = v.x; lds[1] = v.y; lds[2] = v.z; lds[3] = v.w;
}
__device__ __forceinline__ void wait_async0() {}
#endif

// ---------------------------------------------------------------------------
// Tiled f32 WMMA GEMM:  out[n,e] = sum_d In[n,d] * W[e,d]  (+ D[e]*Xaux[n,e])
// Block: 256 threads = 8 waves in 2(M) x 4(N); each wave owns a 32x32 macro
// tile = four 16x16 f32 accumulators (2 A-frags + 2 B-frags -> 4 WMMAs).
// Block tile: 64(M) x 128(N). K staged in 32-wide chunks through a
// double-buffered LDS pipeline fed by GLOBAL_LOAD_ASYNC_TO_LDS_B128.
// ---------------------------------------------------------------------------
__global__ __launch_bounds__(256)
void ssm_gemm_wmma_f32(const float* __restrict__ In,    // [NROWS, DIM]
                       const float* __restrict__ W,     // [DIM, DIM] row-major
                       float*       __restrict__ Out,   // [NROWS, DIM]
                       const float* __restrict__ Dvec,  // [DIM] or null
                       const float* __restrict__ Xaux,  // [NROWS, DIM] or null
                       int*         __restrict__ ctr,   // sync ctrs to zero (mode 0)
                       int nctr, int mode)
{
    __shared__ float As[2][64 * LDT];    // In tile  64 x 32 (stride 36)
    __shared__ float Ws[2][128 * LDT];   // W  tile 128 x 32 (stride 36)

    const int tid  = threadIdx.x;
    const int lane = tid & 31;
    const int wv   = tid >> 5;          // 0..7
    const int m_w  = (wv & 1) * 32;     // wave M offset in block tile
    const int n_w  = (wv >> 1) * 32;    // wave N offset in block tile

    const int m0 = blockIdx.x * 64;     // output row base
    const int e0 = blockIdx.y * 128;    // output col base

    // Phase-1 side job: zero the phase-2 arrival counters (kernel boundary
    // between phase 1 and phase 2 orders these writes before any spin-read).
    if (mode == 0 && ctr) {
        int base = (blockIdx.y * gridDim.x + blockIdx.x) * 8;
        if (tid < 8 && base + tid < nctr) ctr[base + tid] = 0;
    }

    const int mn    = lane & 15;            // M (A) or N (B/D) index
    const int klane = (lane >> 4) << 1;     // K pair base within K=4 group
    const int mhi   = (lane >> 4) << 3;     // +8 rows for upper half-wave (C/D)

    v8f acc00 = {}, acc01 = {}, acc10 = {}, acc11 = {};

    // Stage one K-chunk (64x32 In + 128x32 W) into LDS buffer `buf`.
    auto stage = [&](int buf, int k0) {
        #pragma unroll
        for (int i = 0; i < 2; ++i) {       // 512 float4 of In
            int f = tid + i * 256; int row = f >> 3; int c = (f & 7) * 4;
            async_b128(In + (size_t)(m0 + row) * DIM + k0 + c,
                       &As[buf][row * LDT + c]);
        }
        #pragma unroll
        for (int i = 0; i < 4; ++i) {       // 1024 float4 of W
            int f = tid + i * 256; int row = f >> 3; int c = (f & 7) * 4;
            async_b128(W + (size_t)(e0 + row) * DIM + k0 + c,
                       &Ws[buf][row * LDT + c]);
        }
    };

    stage(0, 0);

    const int NK = DIM / 32;            // 32 K-chunks
    for (int kc = 0; kc < NK; ++kc) {
        const int buf = kc & 1;
        wait_async0();                  // this wave's async copies done
        __syncthreads();                // all waves' copies done / reads done

        if (kc + 1 < NK) stage(buf ^ 1, (kc + 1) * 32);  // overlap next chunk

        #pragma unroll
        for (int kk = 0; kk < 8; ++kk) {
            const int kb = kk * 4 + klane;
            v2f a0, a1, b0, b1;
            a0.x = As[buf][(m_w      + mn) * LDT + kb];
            a0.y = As[buf][(m_w      + mn) * LDT + kb + 1];
            a1.x = As[buf][(m_w + 16 + mn) * LDT + kb];
            a1.y = As[buf][(m_w + 16 + mn) * LDT + kb + 1];
            b0.x = Ws[buf][(n_w      + mn) * LDT + kb];
            b0.y = Ws[buf][(n_w      + mn) * LDT + kb + 1];
            b1.x = Ws[buf][(n_w + 16 + mn) * LDT + kb];
            b1.y = Ws[buf][(n_w + 16 + mn) * LDT + kb + 1];

            acc00 = __builtin_amdgcn_wmma_f32_16x16x4_f32(
                        false, a0, false, b0, (short)0, acc00, false, false);
            acc01 = __builtin_amdgcn_wmma_f32_16x16x4_f32(
                        false, a0, false, b1, (short)0, acc01, false, false);
            acc10 = __builtin_amdgcn_wmma_f32_16x16x4_f32(
                        false, a1, false, b0, (short)0, acc10, false, false);
            acc11 = __builtin_amdgcn_wmma_f32_16x16x4_f32(
                        false, a1, false, b1, (short)0, acc11, false, false);
        }
        __syncthreads();                // reads done before buf is re-staged
    }

    // --- epilogue: C/D layout (VGPR r -> M = r + mhi, lane -> N = mn) ---
    #pragma unroll
    for (int r = 0; r < 8; ++r) {
        const int row0 = m0 + m_w + r + mhi;
        const int row1 = row0 + 16;
        const int col0 = e0 + n_w + mn;
        const int col1 = col0 + 16;
        float v00 = acc00[r], v01 = acc01[r];
        float v10 = acc10[r], v11 = acc11[r];
        if (mode == 1) {    // y = H C^T + D*x
            v00 += Dvec[col0] * Xaux[(size_t)row0 * DIM + col0];
            v01 += Dvec[col1] * Xaux[(size_t)row0 * DIM + col1];
            v10 += Dvec[col0] * Xaux[(size_t)row1 * DIM + col0];
            v11 += Dvec[col1] * Xaux[(size_t)row1 * DIM + col1];
        }
        Out[(size_t)row0 * DIM + col0] = v00;
        Out[(size_t)row0 * DIM + col1] = v01;
        Out[(size_t)row1 * DIM + col0] = v10;
        Out[(size_t)row1 * DIM + col1] = v11;
    }
}

// ---------------------------------------------------------------------------
// Phase 2: recurrence. grid = (16 e-slices, 4 batches), 256 threads.
// Each WG keeps its 64x1024 f32 A-slice resident in LDS (stride 1028 ->
// conflict-free: lane bank = 4r + q + 4j covers all 32 banks). Per step:
// cross-WGP sync via L2 release/acquire counters (cluster barrier fast path),
// h_{t-1} pulled straight into LDS via the async pipe.
// ---------------------------------------------------------------------------
#define LDA 1028   // padded LDS row stride (floats)

__global__ __launch_bounds__(256)
void ssm_recurrence(const float* __restrict__ A,    // [DIM, DIM]
                    const float* __restrict__ bx,   // [BSZ*LSEQ, DIM]
                    float*       __restrict__ H,    // [BSZ*LSEQ, DIM]
                    int*         __restrict__ ctr)  // [BSZ*LSEQ]
{
    extern __shared__ float smem[];
    float* Asl  = smem;                 // 64 * 1028 floats
    float* h_sh = smem + 64 * LDA;      // 1024 floats

    const int tid = threadIdx.x;
    const int e0  = blockIdx.x * 64;    // this WG's output-row slice
    const int b   = blockIdx.y;         // batch
    const int r   = tid >> 2;           // 0..63 : row within slice
    const int q   = tid & 3;            // 0..3  : quarter of the d-dim

    // One-time: load the A slice into LDS (float4, 64 iters/thread).
    for (int i = 0; i < 64; ++i) {
        int f   = i * 256 + tid;        // 0..16383 float4 index
        int row = f >> 8;               // 256 float4 per 1024-float row
        int c   = (f & 255) * 4;
        float4 g = *(const float4*)(A + (size_t)(e0 + row) * DIM + c);
        Asl[row * LDA + c + 0] = g.x; Asl[row * LDA + c + 1] = g.y;
        Asl[row * LDA + c + 2] = g.z; Asl[row * LDA + c + 3] = g.w;
    }
    __syncthreads();

    const size_t bL = (size_t)b * LSEQ;
    const int aoff  = r * LDA + q;

    for (int t = 0; t < LSEQ; ++t) {
        float sum = 0.0f;
        if (t > 0) {
            // Fast path when dispatched as a 16-WG cluster (NOP otherwise).
            __builtin_amdgcn_s_cluster_barrier();
            if (tid == 0) {
                while (__hip_atomic_load(&ctr[bL + t - 1], __ATOMIC_ACQUIRE,
                                         __HIP_MEMORY_SCOPE_AGENT) < 16)
                    __builtin_amdgcn_s_sleep(1);
            }
            __syncthreads();            // also: h_sh reads of step t-1 done
            // Pull full h_{t-1} (4 KB) from L2 directly into LDS.
            async_b128(H + (bL + t - 1) * DIM + tid * 4, h_sh + tid * 4);
            wait_async0();
            __syncthreads();

            #pragma unroll 8
            for (int j = 0; j < 256; ++j)
                sum = fmaf(Asl[aoff + 4 * j], h_sh[q + 4 * j], sum);
        }
        // Reduce the 4 quarter-sums (consecutive wave32 lanes).
        sum += __shfl_xor(sum, 1);
        sum += __shfl_xor(sum, 2);

        const float bxv  = bx[(bL + t) * DIM + e0 + r];
        const float hval = tanhf(sum + bxv);
        if (q == 0) H[(bL + t) * DIM + e0 + r] = hval;

        if (t + 1 < LSEQ)   // warm L2/WGP$ for next step's bx row
            __builtin_prefetch(&bx[(bL + t + 1) * DIM + e0 + r], 0, 0);

        __threadfence();    // publish slice before bumping the counter
        __syncthreads();
        if (tid == 0)
            __hip_atomic_fetch_add(&ctr[bL + t], 1, __ATOMIC_RELEASE,
                                   __HIP_MEMORY_SCOPE_AGENT);
        __syncthreads();
    }
}

// ---------------------------------------------------------------------------
// Host launcher. Workspace: bx (64MB) | H (64MB) | ctr (64KB).
// ---------------------------------------------------------------------------
extern "C" void kernel_launch(void* const* d_in, const int* in_sizes, int n_in,
                              void* d_out, int out_size, void* d_ws, size_t ws_size,
                              hipStream_t stream) {
    const float* x   = (const float*)d_in[0];   // [4,4096,1024]
    const float* A   = (const float*)d_in[1];   // [1024,1024]
    const float* B_w = (const float*)d_in[2];   // [1024,1024]
    const float* C_w = (const float*)d_in[3];   // [1024,1024]
    const float* Dv  = (const float*)d_in[4];   // [1024]
    float* y = (float*)d_out;

    const size_t nhx = (size_t)NROWS * DIM;             // 16,777,216 floats
    float* bx = (float*)d_ws;
    float* H  = bx + nhx;
    int*  ctr = (int*)(H + nhx);
    const int nctr = NROWS;

    dim3 gblk(NROWS / 64, DIM / 128);   // (256, 8)

    // Phase 1: bx = x @ B_w^T  (also zeroes the sync counters)
    ssm_gemm_wmma_f32<<<gblk, 256, 0, stream>>>(x, B_w, bx,
                                                nullptr, nullptr,
                                                ctr, nctr, /*mode=*/0);

    // Phase 2: sequential recurrence, 16 WGs per batch, 261 KB LDS each
    const size_t shmem = (size_t)(64 * LDA + DIM) * sizeof(float);
    ssm_recurrence<<<dim3(16, BSZ), 256, shmem, stream>>>(A, bx, H, ctr);

    // Phase 3: y = H @ C_w^T + D * x
    ssm_gemm_wmma_f32<<<gblk, 256, 0, stream>>>(H, C_w, y,
                                                Dv, x,
                                                nullptr, 0, /*mode=*/1);
}